// NetBackward_74036646248706
// MI455X (gfx1250) — compile-verified
//
#include <hip/hip_runtime.h>

typedef float v2f __attribute__((ext_vector_type(2)));
typedef float v8f __attribute__((ext_vector_type(8)));

#define DIM 512
#define LSTRIDE 36   // 128x32 L tile, padded: 16B-aligned rows, conflict-free frag reads
#define RSTRIDE 132  // 32x128 R tile, padded
#define LTSZ (128 * LSTRIDE)
#define RTSZ (32 * RSTRIDE)

// ---------------------------------------------------------------------------
// Level 0 kernel: dst[j] = (I - src[2j+1]*scale) @ (I - src[2j]*scale).
// Synchronous register staging (the fuse transform must happen in VGPRs).
// ---------------------------------------------------------------------------
__global__ void __launch_bounds__(256)
pair_gemm_fused(const float* __restrict__ src, float* __restrict__ dst, float scale) {
    const int pair = blockIdx.y;
    const int tm = (blockIdx.x & 3) << 7;
    const int tn = (blockIdx.x >> 2) << 7;

    const float* __restrict__ L = src + (size_t)(2 * pair + 1) * (DIM * DIM);
    const float* __restrict__ R = src + (size_t)(2 * pair) * (DIM * DIM);
    float* __restrict__ C = dst + (size_t)pair * (DIM * DIM);

    __shared__ float Lt[LTSZ];
    __shared__ float Rt[RTSZ];

    const int tid = threadIdx.x;
    const int lane = tid & 31;
    const int wave = tid >> 5;
    const int wm = wave & 3;
    const int wn = wave >> 2;
    const int half = lane >> 4;
    const int l15 = lane & 15;

    v8f acc[2][4];
    const v8f vzero = {0.f, 0.f, 0.f, 0.f, 0.f, 0.f, 0.f, 0.f};
#pragma unroll
    for (int i = 0; i < 2; ++i)
#pragma unroll
        for (int j = 0; j < 4; ++j) acc[i][j] = vzero;

    for (int k0 = 0; k0 < DIM; k0 += 32) {
#pragma unroll
        for (int it = 0; it < 4; ++it) {
            int idx = tid + (it << 8);
            int row = idx >> 3;
            int c4 = (idx & 7) << 2;
            float4 v = *(const float4*)(L + (size_t)(tm + row) * DIM + (k0 + c4));
            int gr = tm + row, gc = k0 + c4;
            v.x = ((gr == gc + 0) ? 1.f : 0.f) - v.x * scale;
            v.y = ((gr == gc + 1) ? 1.f : 0.f) - v.y * scale;
            v.z = ((gr == gc + 2) ? 1.f : 0.f) - v.z * scale;
            v.w = ((gr == gc + 3) ? 1.f : 0.f) - v.w * scale;
            *(float4*)(&Lt[row * LSTRIDE + c4]) = v;
        }
#pragma unroll
        for (int it = 0; it < 4; ++it) {
            int idx = tid + (it << 8);
            int row = idx >> 5;
            int c4 = (idx & 31) << 2;
            float4 v = *(const float4*)(R + (size_t)(k0 + row) * DIM + (tn + c4));
            int gr = k0 + row, gc = tn + c4;
            v.x = ((gr == gc + 0) ? 1.f : 0.f) - v.x * scale;
            v.y = ((gr == gc + 1) ? 1.f : 0.f) - v.y * scale;
            v.z = ((gr == gc + 2) ? 1.f : 0.f) - v.z * scale;
            v.w = ((gr == gc + 3) ? 1.f : 0.f) - v.w * scale;
            *(float4*)(&Rt[row * RSTRIDE + c4]) = v;
        }

        int nk = k0 + 32;
        if (nk < DIM) {
            __builtin_prefetch(L + (size_t)(tm + (tid >> 3)) * DIM + nk + ((tid & 7) << 2), 0, 3);
            __builtin_prefetch(R + (size_t)(nk + (tid >> 5)) * DIM + tn + ((tid & 31) << 2), 0, 3);
        }

        __syncthreads();
#pragma unroll
        for (int kk = 0; kk < 32; kk += 4) {
            v2f af[2];
#pragma unroll
            for (int i = 0; i < 2; ++i) {
                int m = (wm << 5) + (i << 4) + l15;
                const float2 t = *(const float2*)(&Lt[m * LSTRIDE + kk + 2 * half]);
                v2f a; a.x = t.x; a.y = t.y;
                af[i] = a;
            }
            v2f bf[4];
#pragma unroll
            for (int j = 0; j < 4; ++j) {
                int nn = (wn << 6) + (j << 4) + l15;
                int kr = kk + 2 * half;
                v2f b; b.x = Rt[kr * RSTRIDE + nn]; b.y = Rt[(kr + 1) * RSTRIDE + nn];
                bf[j] = b;
            }
#pragma unroll
            for (int i = 0; i < 2; ++i)
#pragma unroll
                for (int j = 0; j < 4; ++j)
                    acc[i][j] = __builtin_amdgcn_wmma_f32_16x16x4_f32(
                        false, af[i], false, bf[j], (short)0, acc[i][j], false, false);
        }
        __syncthreads();
    }

#pragma unroll
    for (int i = 0; i < 2; ++i)
#pragma unroll
        for (int j = 0; j < 4; ++j) {
            int base_m = tm + (wm << 5) + (i << 4) + 8 * half;
            int base_n = tn + (wn << 6) + (j << 4) + l15;
#pragma unroll
            for (int r = 0; r < 8; ++r)
                C[(size_t)(base_m + r) * DIM + base_n] = acc[i][j][r];
        }
}

// ---------------------------------------------------------------------------
// Levels 1+: dst[j] = src[2j+1] @ src[2j], double-buffered LDS with
// GLOBAL_LOAD_ASYNC_TO_LDS_B128 (ASYNCcnt-pipelined, no VGPR bounce).
// ---------------------------------------------------------------------------
__device__ __forceinline__ void async_b128(unsigned lds_byte, int gbyte, const char* gbase) {
    asm volatile("global_load_async_to_lds_b128 %0, %1, %2"
                 :: "v"(lds_byte), "v"(gbyte), "s"(gbase)
                 : "memory");
}
__device__ __forceinline__ void wait_async_le(int) {}  // (doc helper)

__global__ void __launch_bounds__(256)
pair_gemm_async(const float* __restrict__ src, float* __restrict__ dst) {
    const int pair = blockIdx.y;
    const int tm = (blockIdx.x & 3) << 7;
    const int tn = (blockIdx.x >> 2) << 7;

    const char* Lb = (const char*)(src + (size_t)(2 * pair + 1) * (DIM * DIM));
    const char* Rb = (const char*)(src + (size_t)(2 * pair) * (DIM * DIM));
    float* __restrict__ C = dst + (size_t)pair * (DIM * DIM);

    __shared__ float Lt[2][LTSZ];
    __shared__ float Rt[2][RTSZ];

    const int tid = threadIdx.x;
    const int lane = tid & 31;
    const int wave = tid >> 5;
    const int wm = wave & 3;
    const int wn = wave >> 2;
    const int half = lane >> 4;
    const int l15 = lane & 15;

    // Per-thread staging coordinates (4 float4 slices of each tile).
    int lrow[4], lc4[4], rrow[4], rc4[4];
#pragma unroll
    for (int it = 0; it < 4; ++it) {
        int idx = tid + (it << 8);
        lrow[it] = idx >> 3;
        lc4[it] = (idx & 7) << 2;
        rrow[it] = idx >> 5;
        rc4[it] = (idx & 31) << 2;
    }

    v8f acc[2][4];
    const v8f vzero = {0.f, 0.f, 0.f, 0.f, 0.f, 0.f, 0.f, 0.f};
#pragma unroll
    for (int i = 0; i < 2; ++i)
#pragma unroll
        for (int j = 0; j < 4; ++j) acc[i][j] = vzero;

    // Issue one K-stage of async copies into buffer b (8 b128 per wave).
    auto issue = [&](int k0, int b) {
#pragma unroll
        for (int it = 0; it < 4; ++it) {
            unsigned lds = (unsigned)(uintptr_t)&Lt[b][lrow[it] * LSTRIDE + lc4[it]];
            int g = ((tm + lrow[it]) * DIM + (k0 + lc4[it])) * 4;
            async_b128(lds, g, Lb);
        }
#pragma unroll
        for (int it = 0; it < 4; ++it) {
            unsigned lds = (unsigned)(uintptr_t)&Rt[b][rrow[it] * RSTRIDE + rc4[it]];
            int g = ((k0 + rrow[it]) * DIM + (tn + rc4[it])) * 4;
            async_b128(lds, g, Rb);
        }
    };

    issue(0, 0);  // prologue: stage 0 -> buffer 0

    const int NS = DIM / 32;  // 16 K-stages
    for (int s = 0; s < NS; ++s) {
        const int cur = s & 1;
        if (s + 1 < NS) {
            issue((s + 1) * 32, cur ^ 1);
            // 8 newer ops in flight; wait until the 8 older (stage s) landed.
            asm volatile("s_wait_asynccnt 0x8" ::: "memory");
        } else {
            asm volatile("s_wait_asynccnt 0x0" ::: "memory");
        }
        __syncthreads();

        const float* __restrict__ Ls = &Lt[cur][0];
        const float* __restrict__ Rs = &Rt[cur][0];
#pragma unroll
        for (int kk = 0; kk < 32; kk += 4) {
            v2f af[2];
#pragma unroll
            for (int i = 0; i < 2; ++i) {
                int m = (wm << 5) + (i << 4) + l15;
                const float2 t = *(const float2*)(&Ls[m * LSTRIDE + kk + 2 * half]);
                v2f a; a.x = t.x; a.y = t.y;
                af[i] = a;
            }
            v2f bf[4];
#pragma unroll
            for (int j = 0; j < 4; ++j) {
                int nn = (wn << 6) + (j << 4) + l15;
                int kr = kk + 2 * half;
                v2f b; b.x = Rs[kr * RSTRIDE + nn]; b.y = Rs[(kr + 1) * RSTRIDE + nn];
                bf[j] = b;
            }
#pragma unroll
            for (int i = 0; i < 2; ++i)
#pragma unroll
                for (int j = 0; j < 4; ++j)
                    acc[i][j] = __builtin_amdgcn_wmma_f32_16x16x4_f32(
                        false, af[i], false, bf[j], (short)0, acc[i][j], false, false);
        }
        __syncthreads();  // all waves done reading buf[cur] before it is re-filled
    }

#pragma unroll
    for (int i = 0; i < 2; ++i)
#pragma unroll
        for (int j = 0; j < 4; ++j) {
            int base_m = tm + (wm << 5) + (i << 4) + 8 * half;
            int base_n = tn + (wn << 6) + (j << 4) + l15;
#pragma unroll
            for (int r = 0; r < 8; ++r)
                C[(size_t)(base_m + r) * DIM + base_n] = acc[i][j][r];
        }
}

// Pass-through (or fused I - A*scale) copy of one matrix, for odd level counts.
__global__ void __launch_bounds__(256)
mat_copy(const float* __restrict__ src, float* __restrict__ dst, float scale, int fuse) {
    int i = blockIdx.x * 256 + threadIdx.x;
    float v = src[i];
    if (fuse) {
        int r = i >> 9, c = i & 511;
        v = (r == c ? 1.0f : 0.0f) - v * scale;
    }
    dst[i] = v;
}

extern "C" void kernel_launch(void* const* d_in, const int* in_sizes, int n_in,
                              void* d_out, int out_size, void* d_ws, size_t ws_size,
                              hipStream_t stream) {
    const float* A = (const float*)d_in[0];
    const int D2 = DIM * DIM;
    const int n = in_sizes[0] / D2;  // 1000
    const float scale = 1.0f / (float)n;

    float* buf0 = (float*)d_ws;
    float* buf1 = buf0 + (size_t)((n + 1) / 2) * D2;
    float* out = (float*)d_out;

    dim3 blk(256);

    if (n == 1) {
        mat_copy<<<dim3(1024), blk, 0, stream>>>(A, out, scale, 1);
        return;
    }

    int count = n;
    const float* src = A;
    bool fused = true;
    int which = 0;
    while (count > 1) {
        int pairs = count >> 1;
        int next = (count + 1) >> 1;
        float* dst = (next == 1) ? out : (which ? buf1 : buf0);
        dim3 grid(16, pairs);
        if (fused)
            pair_gemm_fused<<<grid, blk, 0, stream>>>(src, dst, scale);
        else
            pair_gemm_async<<<grid, blk, 0, stream>>>(src, dst);
        if (count & 1) {
            mat_copy<<<dim3(1024), blk, 0, stream>>>(src + (size_t)(count - 1) * D2,
                                                     dst + (size_t)pairs * D2,
                                                     fused ? scale : 0.0f, fused ? 1 : 0);
        }
        src = dst;
        fused = false;
        which ^= 1;
        count = next;
    }
}